// UnionRGCNLayer_23759759082191
// MI455X (gfx1250) — compile-verified
//
#include <hip/hip_runtime.h>

typedef __attribute__((ext_vector_type(2))) float v2f;
typedef __attribute__((ext_vector_type(8))) float v8f;

#define NN 40000
#define NE 640000
#define NR 500
#define DD 128
#define PP 3
#define MT 4   // row tiles per wave (M blocking): 40000 / (16*4) = 625 blocks exact

// ---------------------------------------------------------------------------
// Zero workspace: S [NN*DD floats] + in_deg [NN ints]
// ---------------------------------------------------------------------------
__global__ __launch_bounds__(256) void k_zero(int* __restrict__ ws, int nwords) {
  int i = blockIdx.x * blockDim.x + threadIdx.x;
  if (i < nwords) ws[i] = 0;
}

// ---------------------------------------------------------------------------
// hW[n,:] = concat(h[n], pos[n]) @ W_hp + b_hp      (N x 131 x 128, fp32 WMMA)
// One wave = one 16-col tile x MT row-tiles; B fragment loaded once per
// k-step and reused across MT A tiles (4x less weight traffic).
// A 16x4 f32 layout: lanes 0-15 hold K={k0,k0+1}, lanes 16-31 K={k0+2,k0+3}.
// ---------------------------------------------------------------------------
__global__ __launch_bounds__(256)
void k_embed(const float* __restrict__ h, const float* __restrict__ pos,
             const float* __restrict__ Whp, const float* __restrict__ bhp,
             float* __restrict__ hW) {
  const int lane = threadIdx.x & 31;
  const int wave = threadIdx.x >> 5;
  const int lmod = lane & 15;
  const int lhi  = lane >> 4;            // 0 or 1
  const int mb = blockIdx.x * 16 * MT;   // 625 blocks * 64 == 40000 exactly
  const int n0 = wave * 16;              // 8 waves * 16 == 128 exactly

  v8f c[MT];
#pragma unroll
  for (int t = 0; t < MT; ++t) c[t] = (v8f){0.f,0.f,0.f,0.f,0.f,0.f,0.f,0.f};

  // main K loop over h columns (0..127)
  for (int k0 = 0; k0 < DD; k0 += 4) {
    const int ka = k0 + 2 * lhi;
    v2f b;
    b.x = Whp[(ka + 0) * DD + n0 + lmod];
    b.y = Whp[(ka + 1) * DD + n0 + lmod];
#pragma unroll
    for (int t = 0; t < MT; ++t) {
      const int arow = mb + 16 * t + lmod;
      v2f a;
      a.x = h[arow * DD + ka];
      a.y = h[arow * DD + ka + 1];
      c[t] = __builtin_amdgcn_wmma_f32_16x16x4_f32(false, a, false, b,
                                                   (short)0, c[t], false, false);
    }
  }
  // tail: pos columns 128..130, zero-padded to K=4
  {
    const int ka = 2 * lhi;              // 0 or 2 within the pos block
    v2f b;
    b.x = Whp[(DD + ka) * DD + n0 + lmod];
    b.y = (ka + 1 < PP) ? Whp[(DD + ka + 1) * DD + n0 + lmod] : 0.f;
#pragma unroll
    for (int t = 0; t < MT; ++t) {
      const int arow = mb + 16 * t + lmod;
      v2f a;
      a.x = pos[arow * PP + ka];
      a.y = (ka + 1 < PP) ? pos[arow * PP + ka + 1] : 0.f;
      c[t] = __builtin_amdgcn_wmma_f32_16x16x4_f32(false, a, false, b,
                                                   (short)0, c[t], false, false);
    }
  }

  const float bias = bhp[n0 + lmod];
#pragma unroll
  for (int t = 0; t < MT; ++t)
#pragma unroll
    for (int i = 0; i < 8; ++i) {
      const int row = mb + 16 * t + i + 8 * lhi;
      hW[row * DD + n0 + lmod] = c[t][i] + bias;
    }
}

// ---------------------------------------------------------------------------
// Edge scatter: S[dst] += hW[src] + emb_rel[type]; in_deg[dst] += 1.
// One wave per edge; each lane handles a float4 chunk (128 = 32 lanes * 4).
// All node arrays are L2-resident (20.5 MB each << 192 MB L2).
// ---------------------------------------------------------------------------
__global__ __launch_bounds__(256)
void k_scatter(const float* __restrict__ hW, const float* __restrict__ rel,
               const int* __restrict__ esrc, const int* __restrict__ edst,
               const int* __restrict__ etyp,
               float* __restrict__ S, int* __restrict__ indeg) {
  const int t = blockIdx.x * blockDim.x + threadIdx.x;
  const int e = t >> 5;
  if (e >= NE) return;
  const int lane = t & 31;
  const int src = esrc[e];
  const int dst = edst[e];
  const int typ = etyp[e];

  const float4 a = ((const float4*)(hW + (size_t)src * DD))[lane];
  const float4 r = ((const float4*)(rel + (size_t)typ * DD))[lane];
  float* out = S + (size_t)dst * DD + lane * 4;
  __hip_atomic_fetch_add(out + 0, a.x + r.x, __ATOMIC_RELAXED, __HIP_MEMORY_SCOPE_AGENT);
  __hip_atomic_fetch_add(out + 1, a.y + r.y, __ATOMIC_RELAXED, __HIP_MEMORY_SCOPE_AGENT);
  __hip_atomic_fetch_add(out + 2, a.z + r.z, __ATOMIC_RELAXED, __HIP_MEMORY_SCOPE_AGENT);
  __hip_atomic_fetch_add(out + 3, a.w + r.w, __ATOMIC_RELAXED, __HIP_MEMORY_SCOPE_AGENT);
  if (lane == 0)
    __hip_atomic_fetch_add(indeg + dst, 1, __ATOMIC_RELAXED, __HIP_MEMORY_SCOPE_AGENT);
}

// ---------------------------------------------------------------------------
// node_repr = relu( (S @ W_n) * norm + h @ (has_in ? W_loop : W_evolve) )
// 3 B matrices loaded once per k-step, reused across MT row tiles:
// 3*MT = 12 chained fp32 WMMAs per k-step.
// ---------------------------------------------------------------------------
__global__ __launch_bounds__(256)
void k_node_out(const float* __restrict__ S, const float* __restrict__ h,
                const float* __restrict__ Wn, const float* __restrict__ Wl,
                const float* __restrict__ Wel, const float* __restrict__ norm,
                const int* __restrict__ indeg, float* __restrict__ out) {
  const int lane = threadIdx.x & 31;
  const int wave = threadIdx.x >> 5;
  const int lmod = lane & 15;
  const int lhi  = lane >> 4;
  const int mb = blockIdx.x * 16 * MT;
  const int n0 = wave * 16;

  v8f c0[MT], c1[MT], c2[MT];
#pragma unroll
  for (int t = 0; t < MT; ++t) {
    c0[t] = (v8f){0.f,0.f,0.f,0.f,0.f,0.f,0.f,0.f};
    c1[t] = c0[t];
    c2[t] = c0[t];
  }

  for (int k0 = 0; k0 < DD; k0 += 4) {
    const int ka = k0 + 2 * lhi;
    v2f bn, bl, be;
    bn.x = Wn [(ka + 0) * DD + n0 + lmod];  bn.y = Wn [(ka + 1) * DD + n0 + lmod];
    bl.x = Wl [(ka + 0) * DD + n0 + lmod];  bl.y = Wl [(ka + 1) * DD + n0 + lmod];
    be.x = Wel[(ka + 0) * DD + n0 + lmod];  be.y = Wel[(ka + 1) * DD + n0 + lmod];
#pragma unroll
    for (int t = 0; t < MT; ++t) {
      const int arow = mb + 16 * t + lmod;
      v2f aS, ah;
      aS.x = S[arow * DD + ka];  aS.y = S[arow * DD + ka + 1];
      ah.x = h[arow * DD + ka];  ah.y = h[arow * DD + ka + 1];
      c0[t] = __builtin_amdgcn_wmma_f32_16x16x4_f32(false, aS, false, bn,
                                                    (short)0, c0[t], false, false);
      c1[t] = __builtin_amdgcn_wmma_f32_16x16x4_f32(false, ah, false, bl,
                                                    (short)0, c1[t], false, false);
      c2[t] = __builtin_amdgcn_wmma_f32_16x16x4_f32(false, ah, false, be,
                                                    (short)0, c2[t], false, false);
    }
  }

#pragma unroll
  for (int t = 0; t < MT; ++t)
#pragma unroll
    for (int i = 0; i < 8; ++i) {
      const int row = mb + 16 * t + i + 8 * lhi;
      const float nm = norm[row];
      const bool has = indeg[row] > 0;
      const float v = c0[t][i] * nm + (has ? c1[t][i] : c2[t][i]);
      out[row * DD + n0 + lmod] = fmaxf(v, 0.f);
    }
}

// ---------------------------------------------------------------------------
// pos_out = relu(pos + pos @ (has_in ? Lp : ELp))   (P=3, trivial VALU)
// ---------------------------------------------------------------------------
__global__ __launch_bounds__(256)
void k_pos_out(const float* __restrict__ pos, const float* __restrict__ Lp,
               const float* __restrict__ ELp, const int* __restrict__ indeg,
               float* __restrict__ out) {
  const int n = blockIdx.x * blockDim.x + threadIdx.x;
  if (n >= NN) return;
  const float* W = (indeg[n] > 0) ? Lp : ELp;
  const float p0 = pos[n * PP + 0];
  const float p1 = pos[n * PP + 1];
  const float p2 = pos[n * PP + 2];
#pragma unroll
  for (int j = 0; j < PP; ++j) {
    const float v = pos[n * PP + j] +
                    p0 * W[0 * PP + j] + p1 * W[1 * PP + j] + p2 * W[2 * PP + j];
    out[n * PP + j] = fmaxf(v, 0.f);
  }
}

// ---------------------------------------------------------------------------
extern "C" void kernel_launch(void* const* d_in, const int* in_sizes, int n_in,
                              void* d_out, int out_size, void* d_ws, size_t ws_size,
                              hipStream_t stream) {
  const float* h    = (const float*)d_in[0];
  const float* pos  = (const float*)d_in[1];
  const float* norm = (const float*)d_in[2];
  const float* rel  = (const float*)d_in[3];
  const float* Whp  = (const float*)d_in[4];
  const float* bhp  = (const float*)d_in[5];
  const float* Wn   = (const float*)d_in[6];
  const float* Wl   = (const float*)d_in[7];
  const float* Wel  = (const float*)d_in[8];
  const float* Lp   = (const float*)d_in[9];
  const float* ELp  = (const float*)d_in[10];
  const int* esrc   = (const int*)d_in[11];
  const int* edst   = (const int*)d_in[12];
  const int* etyp   = (const int*)d_in[13];

  float* out_node = (float*)d_out;                    // [NN*DD]
  float* out_pos  = (float*)d_out + (size_t)NN * DD;  // [NN*PP]

  float* S     = (float*)d_ws;                   // [NN*DD]
  int*   indeg = (int*)d_ws + (size_t)NN * DD;   // [NN]
  float* hW    = out_node;                       // stage hW in d_out, overwritten later

  const int nwords = NN * DD + NN;
  k_zero<<<(nwords + 255) / 256, 256, 0, stream>>>((int*)d_ws, nwords);

  k_embed<<<NN / (16 * MT), 256, 0, stream>>>(h, pos, Whp, bhp, hW);

  k_scatter<<<(NE * 32) / 256, 256, 0, stream>>>(hW, rel, esrc, edst, etyp, S, indeg);

  k_node_out<<<NN / (16 * MT), 256, 0, stream>>>(S, h, Wn, Wl, Wel, norm, indeg, out_node);

  k_pos_out<<<(NN + 255) / 256, 256, 0, stream>>>(pos, Lp, ELp, indeg, out_pos);
}